// SVConvTranspose2d_62989990363671
// MI455X (gfx1250) — compile-verified
//
#include <hip/hip_runtime.h>
#include <stdint.h>

// MI455X (gfx1250) spatially-varying transposed conv.
// BW-bound: weight = 419MB fp32 streamed exactly once (~18us floor @23.3TB/s).
// Per-point GEMM (M=16[4 batch rows used] x N=16[cout] x K=400[tap*cin]) via
// chained V_WMMA_F32_16X16X4_F32. Weight tiles are streamed with
// GLOBAL_LOAD_ASYNC_TO_LDS_B32 (ASYNCcnt, GV form: 64-bit VGPR address) into
// double-buffered LDS, scattered directly into the wave32 WMMA pair layout,
// overlapping HBM streaming with WMMA work.

typedef float v2f __attribute__((ext_vector_type(2)));
typedef float v8f __attribute__((ext_vector_type(8)));

namespace {
constexpr int kH = 128, kW = 128;
// weight strides in floats: [i][o][kh][kw][h][w]
constexpr int wSO  = 5 * 5 * 128 * 128;   // 409600  (cout stride)
constexpr int wSI  = 16 * wSO;            // 6553600 (cin stride)
constexpr int wSKh = 5 * 128 * 128;       // 81920
constexpr int wSKw = 128 * 128;           // 16384
// x / out strides in floats: [n][c][h][w]
constexpr int xSN = 16 * 128 * 128;       // 262144
constexpr int xSI = 128 * 128;            // 16384

constexpr int TILE   = 16;                // output pixels (ow) per workgroup
constexpr int HALO_W = TILE + 4;          // 20
constexpr int HALO_H = 5;
// LDS layouts in float2 (8B) "pair" units; pairs hold (i=2j, i=2j+1)
constexpr int XJ  = HALO_H * HALO_W;      // 100  (j stride in x tile)
constexpr int XN  = 8 * XJ + 1;           // 801  (nb stride, padded)
constexpr int XSZ = 4 * XN;               // 3204 pairs = 25632 B
constexpr int WO  = 8 * TILE + 1;         // 129  (o stride, padded)
constexpr int WSZ = 16 * WO;              // 2064 pairs = 16512 B per buffer
}  // namespace

__global__ void __launch_bounds__(64)
svct_wmma_async_kernel(const float* __restrict__ x, const float* __restrict__ wgt,
                       const float* __restrict__ bias, float* __restrict__ out) {
  __shared__ v2f xls[XSZ];
  __shared__ v2f wls[2 * WSZ];   // double-buffered weight tiles

  const int tid  = threadIdx.x;  // 0..63 (2 waves)
  const int lane = tid & 31;
  const int wid  = tid >> 5;     // wave id 0..1
  const int oh   = blockIdx.y;
  const int ow0  = blockIdx.x * TILE;

  // ---- stage x halo once: x[nb, i, oh-2..oh+2, ow0-2..ow0+17], zero-padded ----
  {
    float* xlf = (float*)xls;
    for (int e = tid; e < 4 * 16 * HALO_H * HALO_W; e += 64) {
      int ww = e % HALO_W;
      int r  = e / HALO_W;
      int hh = r % HALO_H;
      int r2 = r / HALO_H;
      int ci = r2 & 15;
      int nb = r2 >> 4;
      int hg = oh - 2 + hh;
      int wg = ow0 - 2 + ww;
      float v = 0.0f;
      if (hg >= 0 && hg < kH && wg >= 0 && wg < kW)
        v = x[nb * xSN + ci * xSI + hg * kW + wg];
      xlf[2 * (nb * XN + (ci >> 1) * XJ + hh * HALO_W + ww) + (ci & 1)] = v;
    }
  }

  // raw LDS byte address of the weight buffers (LDS aperture: addr[31:0])
  const uint32_t wl_lds = (uint32_t)(uintptr_t)&wls[0];

  // issue one tap's weight tile (256 (i,o) lines x 16 floats) as async->LDS B32
  auto issue_tap = [&](int t, int par) {
    int kh = t / 5, kw = t % 5;
    int hg = oh + 2 - kh;
    int hc = hg < 0 ? 0 : (hg > 127 ? 127 : hg);
    const float* wt = wgt + (size_t)kh * wSKh + (size_t)kw * wSKw +
                      (size_t)hc * kW;
    int wgl = ow0 + (tid & 15) + 2 - kw;
    int wc  = wgl < 0 ? 0 : (wgl > 127 ? 127 : wgl);   // clamped; x==0 kills OOB
    uint32_t base = wl_lds + (uint32_t)par * (WSZ * 8);
    for (int l = 0; l < 64; ++l) {
      int line = l * 4 + (tid >> 4);                   // 0..255
      int ci = line >> 4;
      int o  = line & 15;
      uint64_t gaddr = (uint64_t)(uintptr_t)(wt + (size_t)(ci * wSI + o * wSO + wc));
      uint32_t laddr = base +
          4u * (uint32_t)(2 * (o * WO + (ci >> 1) * TILE + (tid & 15)) + (ci & 1));
      asm volatile("global_load_async_to_lds_b32 %0, %1, off"
                   :: "v"(laddr), "v"(gaddr)
                   : "memory");
    }
  };

  // per-lane WMMA operand indexing (wave32)
  const int m    = lane & 15;   // C/D row group; rows 0..3 = batch n
  const int nb_l = m & 3;
  const int dl   = lane >> 4;   // K-half select (lanes 16-31 take k=2,3)
  const int o_l  = lane & 15;   // B/D column = cout

  const float bias_v = bias[o_l];

  v8f acc[8];
#pragma unroll
  for (int p = 0; p < 8; ++p) acc[p] = {};

  issue_tap(0, 0);  // prologue

  for (int t = 0; t < 25; ++t) {
    asm volatile("s_wait_asynccnt 0" ::: "memory");  // my tap-t loads landed
    __syncthreads();  // publish tap t; everyone done reading buffer parity(t+1)
    if (t < 24) issue_tap(t + 1, (t + 1) & 1);       // overlap with compute

    int kh = t / 5, kw = t % 5;
    int hh = 4 - kh;                                 // row index inside x halo
    const v2f* wbuf = wls + (t & 1) * WSZ;
    const v2f* xp0  = xls + nb_l * XN + hh * HALO_W;
#pragma unroll
    for (int p = 0; p < 8; ++p) {
      int pl = wid * 8 + p;                          // pixel within tile
      const v2f* xp = xp0 + (pl + 4 - kw);           // halo column for pixel
      const v2f* wp = wbuf + o_l * WO + pl;
#pragma unroll
      for (int c = 0; c < 4; ++c) {
        int j = 2 * c + dl;                          // pair index -> k = 2j,2j+1
        v2f A = xp[j * XJ];                          // ds_load_b64 (x tile)
        v2f B = wp[j * TILE];                        // ds_load_b64 (weights)
        acc[p] = __builtin_amdgcn_wmma_f32_16x16x4_f32(
            false, A, false, B, (short)0, acc[p], false, false);
      }
    }
  }

  // ---- store: lanes 0-15 hold D rows m=0..7 in acc comps; rows 0..3 = batch ----
  if (lane < 16) {
#pragma unroll
    for (int p = 0; p < 8; ++p) {
      int ow = ow0 + wid * 8 + p;
#pragma unroll
      for (int r = 0; r < 4; ++r) {
        out[(long long)r * xSN + o_l * xSI + oh * kW + ow] = acc[p][r] + bias_v;
      }
    }
  }
}

extern "C" void kernel_launch(void* const* d_in, const int* in_sizes, int n_in,
                              void* d_out, int out_size, void* d_ws, size_t ws_size,
                              hipStream_t stream) {
  const float* x    = (const float*)d_in[0];
  const float* wgt  = (const float*)d_in[1];
  const float* bias = (const float*)d_in[2];
  float* out        = (float*)d_out;
  dim3 grid(kW / TILE, kH);   // (8, 128) workgroups
  dim3 block(64);             // 2 waves (wave32)
  hipLaunchKernelGGL(svct_wmma_async_kernel, grid, block, 0, stream,
                     x, wgt, bias, out);
}